// CrossDeformableStateSpaceModel_87995289961136
// MI455X (gfx1250) — compile-verified
//
#include <hip/hip_runtime.h>
#include <hip/hip_bf16.h>
#include <math.h>

// ---------------- problem constants ----------------
#define B_   2
#define C_   64
#define H_   32
#define W_   128
#define L_   (H_*W_)      // 4096
#define T_   (2*L_)       // 8192
#define DI_  128          // D_INNER
#define DS_  16           // D_STATE
#define DTR_ 4            // DT_RANK
#define DBL_LD 48         // padded stride for x_proj output (36 -> 48)

typedef float v2f __attribute__((ext_vector_type(2)));
typedef float v8f __attribute__((ext_vector_type(8)));

__device__ __forceinline__ float sigmoidf_(float x){ return 1.f/(1.f+expf(-x)); }
__device__ __forceinline__ float siluf_(float x){ return x*sigmoidf_(x); }
__device__ __forceinline__ float softplusf_(float x){ return (x>20.f)?x:log1pf(expf(x)); }

// ======================================================================
// Generic fp32 GEMM via V_WMMA_F32_16X16X4_F32.
// Y[M,N] = X[M,K] @ Wm[N,K]^T (+bias) (+silu). One wave = one 16x16 tile.
// No lane-dependent branches: operands load as unconditional b64.
// ======================================================================
__global__ void __launch_bounds__(128)
gemm_wmma_f32(float* __restrict__ Y, int ldy,
              const float* __restrict__ X, int ldx,
              const float* __restrict__ Wm, int ldw,
              const float* __restrict__ bias,
              int M, int N, int K, int act)
{
    const int wavesPerBlk = blockDim.x >> 5;
    const int wave  = blockIdx.x * wavesPerBlk + (threadIdx.x >> 5);
    const int tilesN = N >> 4;
    const int tilesM = M >> 4;
    const int tm = wave / tilesN;
    const int tn = wave % tilesN;
    if (tm >= tilesM) return;                 // wave-uniform: EXEC stays full

    const int lane = threadIdx.x & 31;
    const int m    = lane & 15;               // row (A) / col (B,D)
    const int kh   = lane >> 4;               // K-pair select (0: k0..1, 1: k2..3)

    const float* xrow = X  + (size_t)(tm*16 + m) * ldx + 2*kh;
    const float* wrow = Wm + (size_t)(tn*16 + m) * ldw + 2*kh;

    v8f acc = {0.f,0.f,0.f,0.f,0.f,0.f,0.f,0.f};
    #pragma unroll 4
    for (int k = 0; k < K; k += 4) {
        v2f a = *(const v2f*)(xrow + k);
        v2f b = *(const v2f*)(wrow + k);
        acc = __builtin_amdgcn_wmma_f32_16x16x4_f32(false, a, false, b,
                                                    (short)0, acc, false, false);
    }

    // D layout: VGPR i -> row (i + 8*kh), col (lane&15)
    const int col = tn*16 + m;
    const float bv = bias ? bias[col] : 0.f;
    #pragma unroll
    for (int i = 0; i < 8; ++i) {
        const int row = tm*16 + i + 8*kh;
        float v = acc[i] + bv;
        if (act == 1) v = siluf_(v);
        Y[(size_t)row * ldy + col] = v;
    }
}

// ---------------- zero-pad x_proj weights (36x128 -> 48x128) ----------------
__global__ void pad_xproj(const float* __restrict__ src, float* __restrict__ dst)
{
    int i = blockIdx.x*blockDim.x + threadIdx.x;
    if (i >= DBL_LD*DI_) return;
    dst[i] = (i < 36*DI_) ? src[i] : 0.f;
}

// ======================================================================
// 3x3 depthwise conv, zero pad 1, + bias, optional SiLU
// ======================================================================
__global__ void dwconv3x3(const float* __restrict__ in, const float* __restrict__ w,
                          const float* __restrict__ bias, float* __restrict__ out, int act)
{
    int i = blockIdx.x*blockDim.x + threadIdx.x;
    if (i >= B_*C_*L_) return;
    int wq = i % W_;
    int h  = (i / W_) % H_;
    int bc = i / L_;
    int c  = bc % C_;
    const float* base = in + (size_t)bc*L_;
    float s = bias[c];
    #pragma unroll
    for (int ky = 0; ky < 3; ++ky) {
        int yy = h + ky - 1;
        if (yy < 0 || yy >= H_) continue;
        #pragma unroll
        for (int kx = 0; kx < 3; ++kx) {
            int xx = wq + kx - 1;
            if (xx < 0 || xx >= W_) continue;
            s += base[yy*W_ + xx] * w[c*9 + ky*3 + kx];
        }
    }
    out[i] = act ? siluf_(s) : s;
}

// ---------------- offset net: channel mean ----------------
__global__ void chan_mean(const float* __restrict__ xo, float* __restrict__ ymean)
{
    int bc = blockIdx.x;                      // B*C blocks
    __shared__ float red[256];
    float s = 0.f;
    for (int i = threadIdx.x; i < L_; i += 256) s += xo[(size_t)bc*L_ + i];
    red[threadIdx.x] = s; __syncthreads();
    for (int o = 128; o; o >>= 1) { if (threadIdx.x < o) red[threadIdx.x] += red[threadIdx.x+o]; __syncthreads(); }
    if (threadIdx.x == 0) ymean[bc] = red[0] * (1.f/L_);
}

// ---------------- offset net: SE gate (fc1-relu-fc2-sigmoid) ----------------
__global__ void se_gate(const float* __restrict__ ymean,
                        const float* __restrict__ w1, const float* __restrict__ b1,
                        const float* __restrict__ w2, const float* __restrict__ b2,
                        float* __restrict__ se)
{
    int b = blockIdx.x, t = threadIdx.x;      // 64 threads
    __shared__ float hid[4];
    if (t < 4) {
        float s = b1[t];
        for (int c = 0; c < C_; ++c) s += ymean[b*C_ + c] * w1[t*C_ + c];
        hid[t] = fmaxf(s, 0.f);
    }
    __syncthreads();
    float s = b2[t];
    #pragma unroll
    for (int j = 0; j < 4; ++j) s += hid[j] * w2[t*4 + j];
    se[b*C_ + t] = sigmoidf_(s);
}

// ---------------- offset head: SE * x -> gelu -> LN(channel) -> 1x1 conv -> tanh ----------------
__global__ void offset_head(const float* __restrict__ xo, const float* __restrict__ se,
                            const float* __restrict__ lng, const float* __restrict__ lnb,
                            const float* __restrict__ ocw,
                            float* __restrict__ dp, float* __restrict__ dtoff)
{
    int i = blockIdx.x*blockDim.x + threadIdx.x;
    if (i >= B_*L_) return;
    int b = i / L_, l = i % L_;
    float v[C_];
    float mean = 0.f;
    for (int c = 0; c < C_; ++c) {
        float x = xo[((size_t)b*C_ + c)*L_ + l] * se[b*C_ + c];
        x = 0.5f * x * (1.f + erff(x * 0.70710678118654752f));   // exact GELU
        v[c] = x; mean += x;
    }
    mean *= (1.f/C_);
    float var = 0.f;
    for (int c = 0; c < C_; ++c) { float d = v[c]-mean; var += d*d; }
    var *= (1.f/C_);
    float inv = rsqrtf(var + 1e-6f);
    float d0 = 0.f, d1 = 0.f, d2 = 0.f;
    for (int c = 0; c < C_; ++c) {
        float xn = (v[c]-mean)*inv*lng[c] + lnb[c];
        d0 += xn * ocw[c];
        d1 += xn * ocw[C_ + c];
        d2 += xn * ocw[2*C_ + c];
    }
    dp[(size_t)b*2*L_ + l]       = tanhf(d0) * (1.f/W_);
    dp[(size_t)b*2*L_ + L_ + l]  = tanhf(d1) * (1.f/H_);
    dtoff[b*L_ + l]              = tanhf(d2);
}

// ---------------- deformable bilinear sample + sort keys ----------------
__global__ void grid_sample_kern(const float* __restrict__ xdw, const float* __restrict__ pdw,
                                 const float* __restrict__ dp, const float* __restrict__ dtoff,
                                 float* __restrict__ xd, float* __restrict__ pd,
                                 float* __restrict__ keys)
{
    int pix = blockIdx.x;                     // B*L blocks, 64 threads = channels
    int c = threadIdx.x;
    int b = pix / L_, l = pix % L_;
    int h = l / W_, wq = l % W_;
    float gx = -1.f + 2.f*wq/(W_-1) + dp[(size_t)b*2*L_ + l];
    float gy = -1.f + 2.f*h /(H_-1) + dp[(size_t)b*2*L_ + L_ + l];
    float ix = ((gx + 1.f)*W_ - 1.f)*0.5f;
    float iy = ((gy + 1.f)*H_ - 1.f)*0.5f;
    float ix0 = floorf(ix), iy0 = floorf(iy);
    float wx = ix - ix0, wy = iy - iy0;
    const float* xb = xdw + ((size_t)b*C_ + c)*L_;
    const float* pb = pdw + ((size_t)b*C_ + c)*L_;
    auto tap = [&](const float* img, float yy, float xx) -> float {
        bool valid = (xx >= 0.f) && (xx < (float)W_) && (yy >= 0.f) && (yy < (float)H_);
        int xi = (int)fminf(fmaxf(xx, 0.f), (float)(W_-1));
        int yi = (int)fminf(fmaxf(yy, 0.f), (float)(H_-1));
        return valid ? img[yi*W_ + xi] : 0.f;
    };
    float w00 = (1.f-wy)*(1.f-wx), w01 = (1.f-wy)*wx, w10 = wy*(1.f-wx), w11 = wy*wx;
    float ax = tap(xb,iy0,ix0)*w00 + tap(xb,iy0,ix0+1.f)*w01
             + tap(xb,iy0+1.f,ix0)*w10 + tap(xb,iy0+1.f,ix0+1.f)*w11;
    float ap = tap(pb,iy0,ix0)*w00 + tap(pb,iy0,ix0+1.f)*w01
             + tap(pb,iy0+1.f,ix0)*w10 + tap(pb,iy0+1.f,ix0+1.f)*w11;
    xd[((size_t)b*C_ + c)*L_ + l] = ax;
    pd[((size_t)b*C_ + c)*L_ + l] = ap;
    if (c == 0)
        keys[b*L_ + l] = (float)l/(L_-1)*2.f - 1.f + dtoff[b*L_ + l];
}

// ---------------- bitonic argsort over 4096 keys (one workgroup per batch) ----------------
__global__ void __launch_bounds__(1024)
argsort_bitonic(const float* __restrict__ keys, int* __restrict__ order)
{
    __shared__ float k[L_];
    __shared__ int  id[L_];
    int b = blockIdx.x, tid = threadIdx.x;
    for (int i = tid; i < L_; i += 1024) { k[i] = keys[b*L_ + i]; id[i] = i; }
    __syncthreads();
    for (int size = 2; size <= L_; size <<= 1) {
        for (int stride = size >> 1; stride > 0; stride >>= 1) {
            for (int t = tid; t < (L_/2); t += 1024) {
                int lo = ((t / stride) * (stride << 1)) + (t % stride);
                int hi = lo + stride;
                bool asc = ((lo & size) == 0);
                float kl = k[lo], kr = k[hi];
                if ((kl > kr) == asc) {
                    k[lo] = kr; k[hi] = kl;
                    int tmp = id[lo]; id[lo] = id[hi]; id[hi] = tmp;
                }
            }
            __syncthreads();
        }
    }
    for (int i = tid; i < L_; i += 1024) order[b*L_ + i] = id[i];
}

// ---------------- build fwd / bwd / deformable interleaved sequences (B,2L,C) ----------------
__global__ void build_seqs(const float* __restrict__ xdw, const float* __restrict__ pdw,
                           const float* __restrict__ xd,  const float* __restrict__ pd,
                           const int* __restrict__ order,
                           float* __restrict__ fwd, float* __restrict__ bwd, float* __restrict__ defs)
{
    int pix = blockIdx.x;                     // B*L blocks, 64 threads
    int c = threadIdx.x;
    int b = pix / L_, l = pix % L_;
    int h = l / W_, wq = l % W_;
    int lrev = (H_-1-h)*W_ + wq;
    size_t cb = ((size_t)b*C_ + c)*L_;
    size_t row = ((size_t)b*T_ + 2*l)*C_ + c;
    fwd[row]       = xdw[cb + l];
    fwd[row + C_]  = pdw[cb + l];
    bwd[row]       = xdw[cb + lrev];
    bwd[row + C_]  = pdw[cb + lrev];
    int o = order[b*L_ + l];
    defs[row]      = xd[cb + o];
    defs[row + C_] = pd[cb + o];
}

// ---------------- mamba: causal depthwise conv (k=4) + SiLU ----------------
__global__ void conv1d_silu(const float* __restrict__ xz, const float* __restrict__ cw,
                            const float* __restrict__ cb, float* __restrict__ u)
{
    int i = blockIdx.x*blockDim.x + threadIdx.x;
    if (i >= B_*T_*DI_) return;
    int d = i % DI_;
    int t = (i / DI_) % T_;
    int b = i / (DI_*T_);
    float s = cb[d];
    #pragma unroll
    for (int j = 0; j < 4; ++j) {
        int tt = t - 3 + j;
        if (tt >= 0) s += xz[((size_t)b*T_ + tt)*256 + d] * cw[d*4 + j];
    }
    u[((size_t)b*T_ + t)*DI_ + d] = siluf_(s);
}

// ---------------- mamba: dt linear (K=4) + softplus ----------------
__global__ void dt_softplus(const float* __restrict__ dbl, const float* __restrict__ dtw,
                            const float* __restrict__ dtb, float* __restrict__ delta)
{
    int i = blockIdx.x*blockDim.x + threadIdx.x;
    if (i >= B_*T_*DI_) return;
    int d = i % DI_;
    size_t r = i / DI_;                       // b*T + t
    const float* dr = dbl + r*DBL_LD;
    float s = dtb[d];
    #pragma unroll
    for (int j = 0; j < DTR_; ++j) s += dr[j] * dtw[d*DTR_ + j];
    delta[r*DI_ + d] = softplusf_(s);
}

// ---------------- mamba: selective scan. wave32 = 2 channels x 16 states ----------------
__global__ void selective_scan(const float* __restrict__ dbl, const float* __restrict__ delta,
                               const float* __restrict__ u, const float* __restrict__ A_log,
                               const float* __restrict__ Dp, float* __restrict__ y)
{
    int wid = blockIdx.x * (blockDim.x >> 5) + (threadIdx.x >> 5);
    if (wid >= B_*(DI_/2)) return;
    int b = wid / (DI_/2);
    int dpair = wid % (DI_/2);
    int lane = threadIdx.x & 31;
    int s = lane & 15;
    int d = dpair*2 + (lane >> 4);
    float A  = -expf(A_log[d*DS_ + s]);
    float Dv = Dp[d];
    float h = 0.f;
    const float* dblb = dbl + (size_t)b*T_*DBL_LD;
    for (int t = 0; t < T_; ++t) {
        size_t r = (size_t)b*T_ + t;
        float dlt = delta[r*DI_ + d];
        float uu  = u[r*DI_ + d];
        float Bm  = dblb[(size_t)t*DBL_LD + DTR_ + s];
        float Cm  = dblb[(size_t)t*DBL_LD + DTR_ + DS_ + s];
        h = expf(dlt * A) * h + dlt * Bm * uu;
        float part = h * Cm;
        part += __shfl_xor(part, 1, 32);
        part += __shfl_xor(part, 2, 32);
        part += __shfl_xor(part, 4, 32);
        part += __shfl_xor(part, 8, 32);
        if (s == 0) y[r*DI_ + d] = part + uu * Dv;
    }
}

// ---------------- mamba: y *= silu(z) (z = xz[..., 128:]) ----------------
__global__ void gate_z(float* __restrict__ y, const float* __restrict__ xz)
{
    int i = blockIdx.x*blockDim.x + threadIdx.x;
    if (i >= B_*T_*DI_) return;
    int d = i % DI_;
    size_t r = i / DI_;
    y[r*DI_ + d] *= siluf_(xz[r*256 + 128 + d]);
}

// ---------------- final: merge/3 * gate -> channel LayerNorm -> (B,C,H,W) ----------------
__global__ void final_merge(const float* __restrict__ f, const float* __restrict__ bo,
                            const float* __restrict__ dq, const float* __restrict__ gate,
                            const float* __restrict__ lng, const float* __restrict__ lnb,
                            float* __restrict__ out)
{
    int i = blockIdx.x*blockDim.x + threadIdx.x;
    if (i >= B_*L_) return;
    int b = i / L_, l = i % L_;
    const float* fr = f  + (size_t)i*C_;
    const float* br = bo + (size_t)i*C_;
    const float* dr = dq + (size_t)i*C_;
    float v[C_];
    float mean = 0.f;
    for (int c = 0; c < C_; ++c) {
        float g = gate[((size_t)b*C_ + c)*L_ + l];
        float m = (fr[c] + br[c] + dr[c]) * (1.f/3.f) * g;
        v[c] = m; mean += m;
    }
    mean *= (1.f/C_);
    float var = 0.f;
    for (int c = 0; c < C_; ++c) { float d = v[c]-mean; var += d*d; }
    var *= (1.f/C_);
    float inv = rsqrtf(var + 1e-5f);
    for (int c = 0; c < C_; ++c)
        out[((size_t)b*C_ + c)*L_ + l] = (v[c]-mean)*inv*lng[c] + lnb[c];
}

// ======================================================================
// Host side
// ======================================================================
enum {
    IN_MS=0, IN_PAN=1, P_PRE_W=2, P_PRE_B=3, P_PRE_PAN_W=4, P_PRE_PAN_B=5,
    P_DW_W=6, P_DW_B=7, P_OFF_DW_W=8, P_OFF_DW_B=9, P_OFF_FC1_W=10, P_OFF_FC1_B=11,
    P_OFF_FC2_W=12, P_OFF_FC2_B=13, P_OFF_LN_G=14, P_OFF_LN_B=15, P_OFF_CONV_W=16,
    FSSM=17, BSSM=26,   // +0 in_proj, +1 conv_w, +2 conv_b, +3 x_proj, +4 dt_w,
                        // +5 dt_b,  +6 A_log, +7 D, +8 out_proj
    P_FC_W=35, P_FC_B=36, P_BC_W=37, P_BC_B=38, P_DC_W=39, P_DC_B=40,
    P_GATE_W=41, P_GATE_B=42, P_LN_G=43, P_LN_B=44
};

static void launch_gemm(hipStream_t s, float* Y, int ldy, const float* X, int ldx,
                        const float* Wm, int ldw, const float* bias,
                        int M, int N, int K, int act)
{
    int tiles = (M/16)*(N/16);
    int blocks = (tiles + 3)/4;               // 4 waves / 128 threads per block
    gemm_wmma_f32<<<blocks, 128, 0, s>>>(Y, ldy, X, ldx, Wm, ldw, bias, M, N, K, act);
}

static void run_mamba(hipStream_t s, const float* seq, void* const* d_in, int base,
                      float* xz, float* u, float* dbl, float* delta, float* ybuf,
                      float* xpw, float* mo)
{
    const float* in_proj = (const float*)d_in[base+0];
    const float* conv_w  = (const float*)d_in[base+1];
    const float* conv_b  = (const float*)d_in[base+2];
    const float* x_proj  = (const float*)d_in[base+3];
    const float* dt_w    = (const float*)d_in[base+4];
    const float* dt_b    = (const float*)d_in[base+5];
    const float* A_log   = (const float*)d_in[base+6];
    const float* Dvec    = (const float*)d_in[base+7];
    const float* out_prj = (const float*)d_in[base+8];

    const int NE = B_*T_*DI_;
    // in_proj: (B*T,64) @ (256,64)^T -> xz (ld 256)
    launch_gemm(s, xz, 256, seq, C_, in_proj, C_, nullptr, B_*T_, 256, C_, 0);
    conv1d_silu<<<(NE+255)/256, 256, 0, s>>>(xz, conv_w, conv_b, u);
    // x_proj padded to 48 rows, then (B*T,128) @ (48,128)^T -> dbl (ld 48)
    pad_xproj<<<(DBL_LD*DI_+255)/256, 256, 0, s>>>(x_proj, xpw);
    launch_gemm(s, dbl, DBL_LD, u, DI_, xpw, DI_, nullptr, B_*T_, DBL_LD, DI_, 0);
    dt_softplus<<<(NE+255)/256, 256, 0, s>>>(dbl, dt_w, dt_b, delta);
    selective_scan<<<32, 128, 0, s>>>(dbl, delta, u, A_log, Dvec, ybuf);
    gate_z<<<(NE+255)/256, 256, 0, s>>>(ybuf, xz);
    // out_proj: (B*T,128) @ (64,128)^T -> mo (B*T,64)
    launch_gemm(s, mo, C_, ybuf, DI_, out_prj, DI_, nullptr, B_*T_, C_, DI_, 0);
}

extern "C" void kernel_launch(void* const* d_in, const int* in_sizes, int n_in,
                              void* d_out, int out_size, void* d_ws, size_t ws_size,
                              hipStream_t stream)
{
    const float* ms  = (const float*)d_in[IN_MS];
    const float* pan = (const float*)d_in[IN_PAN];

    // ---- carve workspace (all sizes even -> 8B alignment preserved) ----
    float* wsf = (float*)d_ws;
    size_t off = 0;
    auto alloc = [&](size_t n) { float* p = wsf + off; off += n; return p; };
    const size_t IMG = (size_t)B_*C_*L_;
    float* ms_l   = alloc(IMG);
    float* pan_l  = alloc(IMG);
    float* x_dw   = alloc(IMG);
    float* pan_dw = alloc(IMG);
    float* gateb  = alloc(IMG);
    float* xo     = alloc(IMG);
    float* ymean  = alloc(B_*C_);
    float* se     = alloc(B_*C_);
    float* dp     = alloc((size_t)B_*2*L_);
    float* dtoff  = alloc((size_t)B_*L_);
    float* xd     = alloc(IMG);
    float* pd     = alloc(IMG);
    float* keys   = alloc((size_t)B_*L_);
    int*   order  = (int*)alloc((size_t)B_*L_);
    float* fwd    = alloc((size_t)B_*T_*C_);
    float* bwd    = alloc((size_t)B_*T_*C_);
    float* defs   = alloc((size_t)B_*T_*C_);
    float* xz     = alloc((size_t)B_*T_*256);
    float* u      = alloc((size_t)B_*T_*DI_);
    float* dbl    = alloc((size_t)B_*T_*DBL_LD);
    float* delta  = alloc((size_t)B_*T_*DI_);
    float* ybuf   = alloc((size_t)B_*T_*DI_);
    float* xpw    = alloc((size_t)DBL_LD*DI_);   // padded x_proj weights
    float* mo_f   = alloc((size_t)B_*T_*C_);
    float* mo_b   = alloc((size_t)B_*T_*C_);
    float* f_out  = alloc(IMG);
    float* b_out  = alloc(IMG);
    float* d_seq  = alloc(IMG);
    (void)ws_size; (void)in_sizes; (void)n_in;

    const int ROWS = B_*C_*H_;                // 4096 rows of width W_=128

    // 1) pre / pre_pan / gate linears over last dim (WMMA)
    launch_gemm(stream, ms_l,  W_, ms,  W_, (const float*)d_in[P_PRE_W],     W_,
                (const float*)d_in[P_PRE_B],     ROWS, W_, W_, 0);
    launch_gemm(stream, pan_l, W_, pan, W_, (const float*)d_in[P_PRE_PAN_W], W_,
                (const float*)d_in[P_PRE_PAN_B], ROWS, W_, W_, 0);
    launch_gemm(stream, gateb, W_, ms,  W_, (const float*)d_in[P_GATE_W],    W_,
                (const float*)d_in[P_GATE_B],    ROWS, W_, W_, 1);

    // 2) shared depthwise conv + SiLU
    const int NIMG = B_*C_*L_;
    dwconv3x3<<<(NIMG+255)/256, 256, 0, stream>>>(ms_l,  (const float*)d_in[P_DW_W],
                (const float*)d_in[P_DW_B], x_dw,   1);
    dwconv3x3<<<(NIMG+255)/256, 256, 0, stream>>>(pan_l, (const float*)d_in[P_DW_W],
                (const float*)d_in[P_DW_B], pan_dw, 1);

    // 3) offset net
    dwconv3x3<<<(NIMG+255)/256, 256, 0, stream>>>(x_dw, (const float*)d_in[P_OFF_DW_W],
                (const float*)d_in[P_OFF_DW_B], xo, 0);
    chan_mean<<<B_*C_, 256, 0, stream>>>(xo, ymean);
    se_gate<<<B_, 64, 0, stream>>>(ymean,
                (const float*)d_in[P_OFF_FC1_W], (const float*)d_in[P_OFF_FC1_B],
                (const float*)d_in[P_OFF_FC2_W], (const float*)d_in[P_OFF_FC2_B], se);
    offset_head<<<(B_*L_+63)/64, 64, 0, stream>>>(xo, se,
                (const float*)d_in[P_OFF_LN_G], (const float*)d_in[P_OFF_LN_B],
                (const float*)d_in[P_OFF_CONV_W], dp, dtoff);

    // 4) deformable sample + keys, 5) argsort, 6) sequences
    grid_sample_kern<<<B_*L_, 64, 0, stream>>>(x_dw, pan_dw, dp, dtoff, xd, pd, keys);
    argsort_bitonic<<<B_, 1024, 0, stream>>>(keys, order);
    build_seqs<<<B_*L_, 64, 0, stream>>>(x_dw, pan_dw, xd, pd, order, fwd, bwd, defs);

    // 7) two mamba blocks (scratch reused; stream-ordered)
    run_mamba(stream, fwd, d_in, FSSM, xz, u, dbl, delta, ybuf, xpw, mo_f);
    run_mamba(stream, bwd, d_in, BSSM, xz, u, dbl, delta, ybuf, xpw, mo_b);

    // 8) compressors: view (B*T,64) as (B*L,128); silu(linear)
    launch_gemm(stream, f_out, C_, mo_f, DI_, (const float*)d_in[P_FC_W], DI_,
                (const float*)d_in[P_FC_B], B_*L_, C_, DI_, 1);
    launch_gemm(stream, b_out, C_, mo_b, DI_, (const float*)d_in[P_BC_W], DI_,
                (const float*)d_in[P_BC_B], B_*L_, C_, DI_, 1);
    launch_gemm(stream, d_seq, C_, defs, DI_, (const float*)d_in[P_DC_W], DI_,
                (const float*)d_in[P_DC_B], B_*L_, C_, DI_, 1);

    // 9) merge, gate, channel LayerNorm
    final_merge<<<(B_*L_+63)/64, 64, 0, stream>>>(f_out, b_out, d_seq, gateb,
                (const float*)d_in[P_LN_G], (const float*)d_in[P_LN_B], (float*)d_out);
    (void)out_size;
}